// SingleTrajectoryPathLoss_53515292508678
// MI455X (gfx1250) — compile-verified
//
#include <hip/hip_runtime.h>
#include <hip/hip_bf16.h>
#include <math.h>

#define PP 96
#define KK 8
#define WPB 4
#define LOG2PI 1.8378770664093453f
#define SW_COST 1.0f

typedef float v2f __attribute__((ext_vector_type(2)));
typedef float v8f __attribute__((ext_vector_type(8)));

// Exact 32-lane f32 sum via v_wmma_f32_16x16x4_f32 (layout-robust: A=ones,
// B slots all = x  =>  every D row = column sums; sum of one row = 2 * total).
__device__ __forceinline__ float wave_sum32_wmma(float x) {
  v2f a;  a[0] = 1.0f; a[1] = 1.0f;
  v2f bm; bm[0] = x;   bm[1] = x;
  v8f c = {};
  c = __builtin_amdgcn_wmma_f32_16x16x4_f32(false, a, false, bm, (short)0, c,
                                            false, false);
  float r = c[0];
#pragma unroll
  for (int off = 1; off < 16; off <<= 1) r += __shfl_xor(r, off, 32);
  return 0.5f * r;
}

__device__ __forceinline__ void async_ld_b128(unsigned lds_off, const void* gaddr) {
  asm volatile("global_load_async_to_lds_b128 %0, %1, off"
               :: "v"(lds_off), "v"(gaddr) : "memory");
}
__device__ __forceinline__ void wait_async0() {
  asm volatile("s_wait_asynccnt 0" ::: "memory");
}

__global__ __launch_bounds__(WPB * 32)
void stpl_main(const float* __restrict__ logits,
               const float* __restrict__ mu,
               const float* __restrict__ sigma,
               const float* __restrict__ rho,
               const float* __restrict__ rmean,
               const float* __restrict__ rstd,
               const float* __restrict__ ytgt,
               float* __restrict__ ws, int Btot) {
  __shared__ float s_score[WPB][PP * KK];   // log_pi - nll
  __shared__ float s_nll[WPB][PP * KK];     // nll_tk
  __shared__ float4 s_stage[WPB][PP * KK / 4]; // logits staging, later mu_sel
  __shared__ unsigned char s_bp[WPB][PP * 24]; // packed (prev_k | prev_s<<3)
  __shared__ unsigned char s_ksel[WPB][PP];

  const int lane = threadIdx.x & 31;
  const int w    = threadIdx.x >> 5;
  const int b    = blockIdx.x * WPB + w;
  if (b >= Btot) return;
  const int k = lane & 7;
  const float NEG_INF = -__builtin_inff();

  // ---- Phase 0: async-copy this row's logits tile (96x8 f32) to LDS ----
  {
    const float4* gsrc = (const float4*)(logits + (size_t)b * (PP * KK));
#pragma unroll
    for (int i = 0; i < 6; ++i) {
      int idx = i * 32 + lane;
      unsigned loff = (unsigned)(unsigned long long)(void*)&s_stage[w][idx];
      async_ld_b128(loff, (const void*)(gsrc + idx));
    }
    wait_async0();
  }
  const float* slog = (const float*)&s_stage[w][0];

  // ---- Phase 1: NLL + log-softmax -> score/nll tables, entropy & usage ----
  float ent_acc = 0.f, usage_acc = 0.f;
  for (int i = 0; i < 24; ++i) {
    int p = i * 4 + (lane >> 3);
    size_t bkp = ((size_t)(b * KK + k)) * PP + p;
    float2 muv = ((const float2*)mu)[bkp];
    float2 sgv = ((const float2*)sigma)[bkp];
    float  rh  = rho[bkp];
    size_t bp2 = (size_t)b * PP + p;
    float2 rm = ((const float2*)rmean)[bp2];
    float2 rs = ((const float2*)rstd)[bp2];
    float2 yv = ((const float2*)ytgt)[bp2];

    float mx = muv.x * rs.x + rm.x;
    float my = muv.y * rs.y + rm.y;
    float sx = fmaxf(fmaxf(sgv.x * rs.x, 0.8f), 1e-6f);
    float sy = fmaxf(fmaxf(sgv.y * rs.y, 0.8f), 1e-6f);
    float r  = fminf(fmaxf(rh, -0.999f), 0.999f);
    float zx = (yv.x - mx) / sx;
    float zy = (yv.y - my) / sy;
    float one = fmaxf(1.0f - r * r, 1e-6f);
    float nll = logf(sx) + logf(sy) + 0.5f * logf(one) + LOG2PI
              + 0.5f * (zx * zx + zy * zy - 2.0f * r * zx * zy) / one;

    // log-softmax over the 8 lanes of this p-group
    float lg = slog[p * 8 + k];
    float m = lg;
    m = fmaxf(m, __shfl_xor(m, 1, 32));
    m = fmaxf(m, __shfl_xor(m, 2, 32));
    m = fmaxf(m, __shfl_xor(m, 4, 32));
    float e = expf(lg - m);
    float ssum = e;
    ssum += __shfl_xor(ssum, 1, 32);
    ssum += __shfl_xor(ssum, 2, 32);
    ssum += __shfl_xor(ssum, 4, 32);
    float logpi = (lg - m) - logf(ssum);
    float pi = e / ssum;
    ent_acc   += pi * logf(pi + 1e-9f);  // = -entropy contribution
    usage_acc += pi;                      // fixed k per lane

    s_nll[w][p * 8 + k]   = nll;
    s_score[w][p * 8 + k] = logpi - nll;
  }

  // ---- Phase 2: limited-switch Viterbi (24 states, one per lane) ----
  int sgrp = lane >> 3;                       // switch level (3 = dummy)
  float dp = (sgrp == 0) ? s_score[w][k] : NEG_INF;
  int sbase = (sgrp == 0) ? 0 : (sgrp - 1) * 8;
  for (int p = 1; p < PP; ++p) {
    float sc   = s_score[w][p * 8 + k];
    float stay = dp + sc;
    float b1v = NEG_INF, b2v = NEG_INF;
    int   b1i = 0,       b2i = 0;
#pragma unroll
    for (int j = 0; j < 8; ++j) {            // top-2 of previous level
      float vj = __shfl(dp, sbase + j, 32);
      if (vj > b1v)      { b2v = b1v; b2i = b1i; b1v = vj; b1i = j; }
      else if (vj > b2v) { b2v = vj;  b2i = j; }
    }
    float ov = (b1i == k) ? b2v : b1v;
    int   oi = (b1i == k) ? b2i : b1i;
    float swv = ov - SW_COST + sc;
    bool use = (sgrp > 0) && (swv > stay);
    float ndp = use ? swv : stay;
    int pk = use ? oi : k;
    int ps = use ? (sgrp - 1) : sgrp;
    if (lane < 24) s_bp[w][p * 24 + lane] = (unsigned char)(pk | (ps << 3));
    dp = ndp;
  }

  // argmax over flat = k*3 + s (ties -> lowest flat, matching jnp.argmax)
  float fv = (lane < 24) ? dp : NEG_INF;
  int   fl = (lane < 24) ? (k * 3 + sgrp) : 0x7fffffff;
#pragma unroll
  for (int off = 16; off >= 1; off >>= 1) {
    float ovv = __shfl_xor(fv, off, 32);
    int   ofl = __shfl_xor(fl, off, 32);
    if (ovv > fv || (ovv == fv && ofl < fl)) { fv = ovv; fl = ofl; }
  }
  int end_k = fl / 3, end_s = fl % 3;

  if (lane == 0) {                           // backtrace
    int kk2 = end_k, ss = end_s;
    s_ksel[w][PP - 1] = (unsigned char)kk2;
    for (int p = PP - 1; p >= 1; --p) {
      unsigned char e = s_bp[w][p * 24 + ss * 8 + kk2];
      kk2 = e & 7; ss = e >> 3;
      s_ksel[w][p - 1] = (unsigned char)kk2;
    }
  }

  // ---- Phase 3: gather selected path terms + smoothness ----
  float path_acc = 0.f, gate_acc = 0.f, smooth_acc = 0.f;
  float2* smu = (float2*)&s_stage[w][0];     // reuse staging LDS
#pragma unroll
  for (int t = 0; t < 3; ++t) {
    int p  = t * 32 + lane;
    int kq = (int)s_ksel[w][p];
    float nl = s_nll[w][p * 8 + kq];
    float sc = s_score[w][p * 8 + kq];
    path_acc += nl;
    gate_acc -= (sc + nl);                   // -log_pi_sel
    size_t bkp = ((size_t)(b * KK + kq)) * PP + p;
    float2 muv = ((const float2*)mu)[bkp];   // L2-resident second pass
    size_t bp2 = (size_t)b * PP + p;
    float2 rm = ((const float2*)rmean)[bp2];
    float2 rs = ((const float2*)rstd)[bp2];
    float2 ms; ms.x = muv.x * rs.x + rm.x; ms.y = muv.y * rs.y + rm.y;
    smu[p] = ms;
  }
#pragma unroll
  for (int t = 0; t < 3; ++t) {
    int p = t * 32 + lane;
    if (p < PP - 1) {
      float2 a0 = smu[p], a1 = smu[p + 1];
      float dx = a1.x - a0.x, dy = a1.y - a0.y;
      smooth_acc += dx * dx + dy * dy;
    }
  }

  // ---- Reductions (WMMA wave sums) + global accumulation ----
  float u = usage_acc;                       // per-k usage: lanes k,k+8,k+16,k+24
  u += __shfl_xor(u, 8, 32);
  u += __shfl_xor(u, 16, 32);
  float pw  = wave_sum32_wmma(path_acc);
  float gw  = wave_sum32_wmma(gate_acc);
  float ew  = wave_sum32_wmma(ent_acc);
  float smw = wave_sum32_wmma(smooth_acc);
  if (lane < 8) atomicAdd(&ws[4 + lane], u);
  if (lane == 0) {
    atomicAdd(&ws[0], pw);
    atomicAdd(&ws[1], gw);
    atomicAdd(&ws[2], ew);
    atomicAdd(&ws[3], smw);
  }
}

__global__ void stpl_zero(float* ws) {
  int i = threadIdx.x;
  if (i < 12) ws[i] = 0.0f;
}

__global__ void stpl_finalize(const float* __restrict__ ws,
                              float* __restrict__ out, int Btot) {
  if (threadIdx.x != 0 || blockIdx.x != 0) return;
  float inv_bp = 1.0f / ((float)Btot * (float)PP);
  float path   = ws[0] * inv_bp;
  float gate   = ws[1] * inv_bp;
  float entl   = ws[2] * inv_bp;                 // mean of -entropy
  float smooth = ws[3] / ((float)Btot * (float)(PP - 1));
  float lb = 0.f;
  for (int kk2 = 0; kk2 < 8; ++kk2) {
    float uu = ws[4 + kk2] * inv_bp;
    float d = uu - 0.125f;
    lb += d * d;
  }
  lb *= (1.0f / 8.0f);
  out[0] = 1.0f * path + 0.1f * gate + 0.01f * entl + 0.01f * lb + 0.1f * smooth;
}

extern "C" void kernel_launch(void* const* d_in, const int* in_sizes, int n_in,
                              void* d_out, int out_size, void* d_ws, size_t ws_size,
                              hipStream_t stream) {
  const float* logits = (const float*)d_in[0];
  const float* mu     = (const float*)d_in[1];
  const float* sigma  = (const float*)d_in[2];
  const float* rho    = (const float*)d_in[3];
  const float* rmean  = (const float*)d_in[4];
  const float* rstd   = (const float*)d_in[5];
  const float* ytgt   = (const float*)d_in[6];
  float* ws  = (float*)d_ws;
  float* out = (float*)d_out;
  int Btot = in_sizes[0] / (PP * KK);
  int blocks = (Btot + WPB - 1) / WPB;
  hipLaunchKernelGGL(stpl_zero, dim3(1), dim3(32), 0, stream, ws);
  hipLaunchKernelGGL(stpl_main, dim3(blocks), dim3(WPB * 32), 0, stream,
                     logits, mu, sigma, rho, rmean, rstd, ytgt, ws, Btot);
  hipLaunchKernelGGL(stpl_finalize, dim3(1), dim3(1), 0, stream, ws, out, Btot);
}